// _System_contractive_37477884625201
// MI455X (gfx1250) — compile-verified
//
#include <hip/hip_runtime.h>
#include <hip/hip_bf16.h>
#include <math.h>

// MI455X / gfx1250, wave32. All heavy math through V_WMMA_F32_16X16X4_F32.
#define NX 512
#define NQ 512
#define NU 128
#define BATCH 32768
#define EPS 1e-3f

typedef float v2f __attribute__((ext_vector_type(2)));
typedef float v8f __attribute__((ext_vector_type(8)));

// ---------------------------------------------------------------------------
// Generic fused GEMM:  Out[m,n] = scale * sum_s sum_k In_s[m,k]*opB(W_s)[k,n]
//                                 + diag_add * (m==n)
// tB==0: W is (N,K) row-major (Out = In * W^T);  tB==1: W is (K,N) row-major.
// Block tile 128x128, 8 waves (4 along M x 2 along N), each wave a 32x64
// tile of 2x4 16x16 fragments: A-frag reused 4x, B-frag 2x per k-step
// (0.1875 B/MAC operand traffic vs 0.25 for a 32x32 tile).
// ---------------------------------------------------------------------------
__device__ inline void gemm_seg(const float* __restrict__ In,
                                const float* __restrict__ W,
                                int K, int tB, int N,
                                int m0, int n0, int lane, v8f c[2][4]) {
  if (K <= 0 || In == nullptr) return;
  const int mr   = m0 + (lane & 15);
  const int nc   = n0 + (lane & 15);
  const int koff = (lane >> 4) * 2;          // lanes 16..31 carry K+2,K+3
  if (tB == 0) {
    for (int k = 0; k < K; k += 4) {
      v2f a[2], b[4];
#pragma unroll
      for (int i = 0; i < 2; ++i)
        a[i] = *(const v2f*)(In + (size_t)(mr + 16 * i) * K + (k + koff));
#pragma unroll
      for (int i = 0; i < 4; ++i)
        b[i] = *(const v2f*)(W + (size_t)(nc + 16 * i) * K + (k + koff));
#pragma unroll
      for (int mi = 0; mi < 2; ++mi)
#pragma unroll
        for (int ni = 0; ni < 4; ++ni)
          c[mi][ni] = __builtin_amdgcn_wmma_f32_16x16x4_f32(
              false, a[mi], false, b[ni], (short)0, c[mi][ni], false, false);
    }
  } else {
    for (int k = 0; k < K; k += 4) {
      v2f a[2], b[4];
#pragma unroll
      for (int i = 0; i < 2; ++i)
        a[i] = *(const v2f*)(In + (size_t)(mr + 16 * i) * K + (k + koff));
#pragma unroll
      for (int i = 0; i < 4; ++i) {
        const float* p = W + (size_t)(k + koff) * N + (nc + 16 * i);
        b[i].x = p[0]; b[i].y = p[N];
      }
#pragma unroll
      for (int mi = 0; mi < 2; ++mi)
#pragma unroll
        for (int ni = 0; ni < 4; ++ni)
          c[mi][ni] = __builtin_amdgcn_wmma_f32_16x16x4_f32(
              false, a[mi], false, b[ni], (short)0, c[mi][ni], false, false);
    }
  }
}

__global__ __launch_bounds__(256) void gemm_xwt(
    const float* __restrict__ In0, const float* __restrict__ W0, int K0, int tB0,
    const float* __restrict__ In1, const float* __restrict__ W1, int K1, int tB1,
    const float* __restrict__ In2, const float* __restrict__ W2, int K2, int tB2,
    float* __restrict__ Out, int M, int N, float scale, float diag_add) {
  const int lane = threadIdx.x & 31;
  const int wid  = threadIdx.x >> 5;
  const int m0 = blockIdx.y * 128 + (wid >> 1) * 32;   // 4 waves along M
  const int n0 = blockIdx.x * 128 + (wid & 1) * 64;    // 2 waves along N
  (void)M;
  v8f c[2][4] = {};
  gemm_seg(In0, W0, K0, tB0, N, m0, n0, lane, c);
  gemm_seg(In1, W1, K1, tB1, N, m0, n0, lane, c);
  gemm_seg(In2, W2, K2, tB2, N, m0, n0, lane, c);
  // C/D layout: VGPR r -> lanes0-15: M=r, lanes16-31: M=r+8; N = lane&15
  const int mhi = (lane >> 4) * 8;
  const int ncl = lane & 15;
#pragma unroll
  for (int mi = 0; mi < 2; ++mi)
#pragma unroll
    for (int ni = 0; ni < 4; ++ni)
#pragma unroll
      for (int r = 0; r < 8; ++r) {
        int m = m0 + mi * 16 + r + mhi;
        int n = n0 + ni * 16 + ncl;
        float v = c[mi][ni][r] * scale + ((m == n) ? diag_add : 0.f);
        Out[(size_t)m * N + n] = v;
      }
}

// ---------------------------------------------------------------------------
// Elementwise prep from H (1024x1024), Y1, Chi: lam, Y, D11, C1, G
// ---------------------------------------------------------------------------
__global__ void prep_elem(const float* __restrict__ H,
                          const float* __restrict__ Y1,
                          const float* __restrict__ Chi,
                          float* __restrict__ lam, float* __restrict__ Ym,
                          float* __restrict__ D11, float* __restrict__ C1,
                          float* __restrict__ G) {
  int idx = blockIdx.x * blockDim.x + threadIdx.x;
  if (idx >= NX * NQ) return;
  int i = idx >> 9, k = idx & 511;
  float lam_i = 0.5f * H[(size_t)(NX + i) * 1024 + NX + i];
  if (k == 0) lam[i] = lam_i;
  float h1 = H[(size_t)i * 1024 + k];
  float h2 = H[(size_t)i * 1024 + NX + k];
  float h4 = H[(size_t)(NX + i) * 1024 + NX + k];
  Ym[idx]  = -0.5f * (h1 + Y1[idx] - Y1[k * NX + i]);   // ALPHA==0
  D11[idx] = (k < i) ? (-h4 / lam_i) : 0.f;             // -tril(H4,-1)/lam
  C1[idx]  = Chi[k * NQ + i] / lam_i;                   // Chi^T / lam
  G[idx]   = -h2 - Chi[idx];                            // -H2 - Chi
}

// ---------------------------------------------------------------------------
// Newton-Schulz seed: s = max row abs-sum of P; X0 = (1/s) * I
// ---------------------------------------------------------------------------
__global__ void rowsum_abs(const float* __restrict__ P, float* __restrict__ rs) {
  int r = blockIdx.x * blockDim.x + threadIdx.x;
  if (r >= NX) return;
  float s = 0.f;
  for (int k = 0; k < NX; ++k) s += fabsf(P[(size_t)r * NX + k]);
  rs[r] = s;
}

__global__ __launch_bounds__(256) void init_X0(const float* __restrict__ rs,
                                               float* __restrict__ X0) {
  __shared__ float red[256];
  int tid = threadIdx.x;
  red[tid] = fmaxf(rs[tid], rs[tid + 256]);
  __syncthreads();
  for (int s = 128; s > 0; s >>= 1) {
    if (tid < s) red[tid] = fmaxf(red[tid], red[tid + s]);
    __syncthreads();
  }
  float c = 1.0f / red[0];
  for (int e = tid; e < NX * NX; e += 256) {
    int r = e >> 9, k = e & 511;
    X0[e] = (r == k) ? c : 0.f;
  }
}

// ---------------------------------------------------------------------------
// Column-sequential tanh recurrence. Each block owns 128 batch rows; the
// growing w-tile (128x512 f32) lives in LDS (~298KB of the 320KB WGP LDS).
// Per 16-column block: WMMA computes S = w_tile * D11_panel^T (history),
// then 128 threads run the 16 serial tanh steps in registers.
// ---------------------------------------------------------------------------
#define W_LD 514   // 512 + pad(2) -> bank spread for row-strided LDS reads

__global__ __launch_bounds__(256) void recur_kernel(
    const float* __restrict__ base, const float* __restrict__ D11,
    float* __restrict__ w_out) {
  extern __shared__ float lds[];
  float* w_s  = lds;                       // 128 x W_LD
  float* dpan = w_s + 128 * W_LD;          // 16  x W_LD
  float* S_s  = dpan + 16 * W_LD;          // 128 x 17
  const int tid  = threadIdx.x;
  const int lane = tid & 31;
  const int wid  = tid >> 5;
  const size_t row0 = (size_t)blockIdx.x * 128;

  for (int j = 0; j < NQ / 16; ++j) {
    const int i0 = j * 16;
    const int Wd = i0 + 16;
    // stage D11 rows [i0,i0+16), cols [0,Wd) into LDS (8x reuse by waves)
    for (int e = tid; e < 16 * Wd; e += 256) {
      int r = e / Wd, cc = e - r * Wd;
      dpan[r * W_LD + cc] = D11[(size_t)(i0 + r) * NQ + cc];
    }
    __syncthreads();

    // S(16x16 per wave) = w_s[wave rows, :i0] @ dpan[:, :i0]^T
    v8f cf = {};
    {
      const int mrow = wid * 16 + (lane & 15);
      const int ncol = lane & 15;
      const int koff = (lane >> 4) * 2;
      for (int k = 0; k < i0; k += 4) {
        v2f a, b;
        a.x = w_s[mrow * W_LD + k + koff];
        a.y = w_s[mrow * W_LD + k + koff + 1];
        b.x = dpan[ncol * W_LD + k + koff];
        b.y = dpan[ncol * W_LD + k + koff + 1];
        cf = __builtin_amdgcn_wmma_f32_16x16x4_f32(false, a, false, b,
                                                   (short)0, cf, false, false);
      }
    }
    {
      const int mhi = (lane >> 4) * 8;
      const int n   = lane & 15;
#pragma unroll
      for (int r = 0; r < 8; ++r)
        S_s[(wid * 16 + r + mhi) * 17 + n] = cf[r];
    }
    __syncthreads();

    // 16 sequential tanh steps; one thread per batch row
    if (tid < 128) {
      float wloc[16], bv[16];
      const size_t gb = (row0 + tid) * NQ + i0;
#pragma unroll
      for (int l = 0; l < 16; ++l) bv[l] = base[gb + l];
#pragma unroll
      for (int l = 0; l < 16; ++l) {
        float v = bv[l] + S_s[tid * 17 + l];
        for (int t = 0; t < l; ++t)
          v += wloc[t] * dpan[l * W_LD + i0 + t];   // uniform broadcast read
        wloc[l] = tanhf(v);
      }
#pragma unroll
      for (int l = 0; l < 16; ++l) w_s[tid * W_LD + i0 + l] = wloc[l];
    }
    __syncthreads();
  }

  // coalesced write-back of the finished w tile
  for (int e = tid; e < 128 * NQ; e += 256) {
    int r = e >> 9, cc = e & 511;
    w_out[(row0 + r) * NQ + cc] = w_s[r * W_LD + cc];
  }
}

// ---------------------------------------------------------------------------
extern "C" void kernel_launch(void* const* d_in, const int* in_sizes, int n_in,
                              void* d_out, int out_size, void* d_ws, size_t ws_size,
                              hipStream_t stream) {
  (void)in_sizes; (void)n_in; (void)out_size; (void)ws_size;
  const float* xi    = (const float*)d_in[1];
  const float* u     = (const float*)d_in[2];
  const float* Pstar = (const float*)d_in[3];
  const float* Chi   = (const float*)d_in[4];
  const float* Y1    = (const float*)d_in[5];
  const float* B2    = (const float*)d_in[6];
  const float* D12   = (const float*)d_in[7];
  const float* Xin   = (const float*)d_in[8];
  float* out = (float*)d_out;

  float* ws  = (float*)d_ws;
  float* H   = ws;  ws += 1024 * 1024;
  float* P   = ws;  ws += NX * NX;
  float* lam = ws;  ws += NX;
  float* Ym  = ws;  ws += NX * NX;
  float* D11 = ws;  ws += NX * NX;
  float* C1  = ws;  ws += NX * NX;
  float* G   = ws;  ws += NX * NX;
  float* Xa  = ws;  ws += NX * NX;
  float* Xb  = ws;  ws += NX * NX;
  float* T   = ws;  ws += NX * NX;
  float* rs  = ws;  ws += NX;
  float* A   = ws;  ws += NX * NX;
  float* B1m = ws;  ws += NX * NX;
  float* w   = ws;  ws += (size_t)BATCH * NQ;
  float* base = out;   // d_out is dead until the final GEMM -> use as scratch

  const float* nullf = nullptr;
  auto gemm = [&](const float* In0, const float* W0, int K0, int tB0,
                  const float* In1, const float* W1, int K1, int tB1,
                  const float* In2, const float* W2, int K2, int tB2,
                  float* Out, int M, int N, float scale, float diag) {
    dim3 g(N / 128, M / 128);
    hipLaunchKernelGGL(gemm_xwt, g, dim3(256), 0, stream,
                       In0, W0, K0, tB0, In1, W1, K1, tB1, In2, W2, K2, tB2,
                       Out, M, N, scale, diag);
  };

  // H = X X^T + eps I ; P = 0.5 Pstar Pstar^T + eps I
  gemm(Xin, Xin, 1024, 0, nullf, nullf, 0, 0, nullf, nullf, 0, 0,
       H, 1024, 1024, 1.f, EPS);
  gemm(Pstar, Pstar, NX, 0, nullf, nullf, 0, 0, nullf, nullf, 0, 0,
       P, NX, NX, 0.5f, EPS);

  hipLaunchKernelGGL(prep_elem, dim3((NX * NQ) / 256), dim3(256), 0, stream,
                     H, Y1, Chi, lam, Ym, D11, C1, G);

  // Pinv via Newton-Schulz: X0 = I/||P||inf, X <- X (2I - P X)
  hipLaunchKernelGGL(rowsum_abs, dim3(2), dim3(256), 0, stream, P, rs);
  hipLaunchKernelGGL(init_X0, dim3(1), dim3(256), 0, stream, rs, Xa);
  float* Xc = Xa; float* Xn = Xb;
  for (int it = 0; it < 24; ++it) {
    gemm(P, Xc, NX, 1, nullf, nullf, 0, 0, nullf, nullf, 0, 0,
         T, NX, NX, -1.f, 2.f);                         // T = 2I - P X
    gemm(Xc, T, NX, 1, nullf, nullf, 0, 0, nullf, nullf, 0, 0,
         Xn, NX, NX, 1.f, 0.f);                         // X' = X T
    float* tmp = Xc; Xc = Xn; Xn = tmp;
  }
  float* Pinv = Xc;

  // A = Pinv Y ; B1 = Pinv (-H2 - Chi)
  gemm(Pinv, Ym, NX, 1, nullf, nullf, 0, 0, nullf, nullf, 0, 0,
       A, NX, NX, 1.f, 0.f);
  gemm(Pinv, G, NX, 1, nullf, nullf, 0, 0, nullf, nullf, 0, 0,
       B1m, NX, NX, 1.f, 0.f);

  // base = xi C1^T + u D12^T   (staged in d_out)
  gemm(xi, C1, NX, 0, u, D12, NU, 0, nullf, nullf, 0, 0,
       base, BATCH, NQ, 1.f, 0.f);

  // sequential tanh recurrence -> w
  size_t smem = (size_t)(128 * W_LD + 16 * W_LD + 128 * 17) * sizeof(float);
  hipLaunchKernelGGL(recur_kernel, dim3(BATCH / 128), dim3(256), smem, stream,
                     base, D11, w);

  // xi_ = xi A^T + w B1^T + u B2^T
  gemm(xi, A, NX, 0, w, B1m, NQ, 0, u, B2, NU, 0,
       out, BATCH, NX, 1.f, 0.f);
}